// DNAShapeNet_75900662055395
// MI455X (gfx1250) — compile-verified
//
#include <hip/hip_runtime.h>

typedef __attribute__((ext_vector_type(2))) float v2f;
typedef __attribute__((ext_vector_type(8))) float v8f;

#define TILE   128
#define STRIDE 160
#define SLEN   8192
#define BATCH  128

struct Params {
  const float* x;
  const float* w[4];
  const float* b[4];
  const float* g[4];
  const float* bb[4];
  const float* rm[4];
  const float* rv[4];
  const float* fw1;
  const float* fb1;
  const float* fw2;
  const float* fb2;
  float* out;
};

// ---------------------------------------------------------------------------
// Weight staging: permute global weights [32][CIN][KW] into WMMA A-fragment
// order in LDS.  K dim flattened as kc = kw*CIN + cin (CIN is a power of 2).
// Layout: wl[(ks*2 + mt)*64 + lane*2 + j] = W[m][kc]  with
//   m = (lane&15) + 16*mt,  kc = 4*ks + 2*(lane>>4) + j.
// A fragment load is then one ds_load_b64 (pairs merge into ds_load_2addr_b64).
// ---------------------------------------------------------------------------
template<int CIN, int KW>
__device__ __forceinline__ void stage_weights(const float* __restrict__ wg,
                                              float* wl, int tid) {
  constexpr int CK  = CIN * KW;
  constexpr int L2C = (CIN == 4) ? 2 : 5;
  constexpr int TOT = CK * 32;                 // (CK/4 ksteps) * 2 mt * 32 lanes * 2
  for (int o = tid; o < TOT; o += 256) {
    const int j    = o & 1;
    const int lane = (o >> 1) & 31;
    const int mt   = (o >> 6) & 1;
    const int ks   = o >> 7;
    const int m    = (lane & 15) + 16 * mt;
    const int kc   = ks * 4 + 2 * (lane >> 4) + j;
    const int cin  = kc & (CIN - 1);
    const int kw   = kc >> L2C;
    wl[o] = wg[(m * CIN + cin) * KW + kw];
  }
}

// ---------------------------------------------------------------------------
// Conv+bias+ReLU+BN as valid conv over a haloed LDS tile via f32 WMMA 16x16x4.
// Fully-unrolled K loop -> every DS access is base-reg + immediate offset.
// ---------------------------------------------------------------------------
template<int CIN, int KW, int NT_OUT>
__device__ __forceinline__ void conv_layer(
    const float* wl,                   // LDS staged weights (fragment order)
    const float* lin,                  // LDS in  [CIN][STRIDE]
    float* lout,                       // LDS out [32][STRIDE]
    const float* bias, const float* scale, const float* shift,  // LDS [32]
    int lane, int wave, int gbase)
{
  constexpr int CK  = CIN * KW;
  constexpr int L2C = (CIN == 4) ? 2 : 5;
  const int n_   = lane & 15;
  const int half = lane >> 4;
  const v2f* aw  = reinterpret_cast<const v2f*>(wl) + lane;
  const float* bi = bias  + 8 * half;          // per-lane constant bases
  const float* sc = scale + 8 * half;
  const float* sh = shift + 8 * half;

  for (int nt = wave; nt < NT_OUT; nt += 8) {
    const int nb = nt * 16;
    const float* b2 = lin + 2 * half * STRIDE + nb + n_;   // per-lane B base
    v8f acc0 = {};
    v8f acc1 = {};
    #pragma unroll
    for (int ks = 0; ks < CK / 4; ++ks) {
      const int kb   = 4 * ks;                 // compile-time after unroll
      const int cinb = kb & (CIN - 1);         // no wrap: cinb + 2*half + 1 <= CIN-1
      const int kwb  = kb >> L2C;
      v2f bf;
      bf.x = b2[cinb * STRIDE + kwb];          // K = kb + 2*half
      bf.y = b2[(cinb + 1) * STRIDE + kwb];    // K = kb + 2*half + 1
      const v2f a0 = aw[(ks * 2 + 0) * 32];    // M rows 0..15
      const v2f a1 = aw[(ks * 2 + 1) * 32];    // M rows 16..31
      acc0 = __builtin_amdgcn_wmma_f32_16x16x4_f32(false, a0, false, bf,
                                                   (short)0, acc0, false, false);
      acc1 = __builtin_amdgcn_wmma_f32_16x16x4_f32(false, a1, false, bf,
                                                   (short)0, acc1, false, false);
    }
    // Epilogue: +bias, ReLU, BN affine; zero outside [0,S) (per-layer zero pad).
    // Per-lane store base -> all DS store offsets are immediates.
    const int gpos = gbase + nb + n_;
    const bool valid = (unsigned)gpos < (unsigned)SLEN;
    float* o2 = lout + 8 * half * STRIDE + nb + n_;
    #pragma unroll
    for (int i = 0; i < 8; ++i) {
      float y0 = fmaxf(acc0[i] + bi[i], 0.0f);
      y0 = y0 * sc[i] + sh[i];
      o2[i * STRIDE] = valid ? y0 : 0.0f;
      float y1 = fmaxf(acc1[i] + bi[i + 16], 0.0f);
      y1 = y1 * sc[i + 16] + sh[i + 16];
      o2[(i + 16) * STRIDE] = valid ? y1 : 0.0f;
    }
  }
}

__global__ __launch_bounds__(256)
void dnashape_fused_kernel(Params p) {
  __shared__ float xbuf[4 * STRIDE];
  __shared__ float bufA[32 * STRIDE];
  __shared__ float bufB[32 * STRIDE];
  __shared__ __align__(16) float wstage[32 * 32 * 7];   // max CK*32 (layer 3)
  __shared__ float cbias[4][32], cscale[4][32], cshift[4][32];
  __shared__ float fb1s[16], fw2s[16], fb2s[1];

  const int tid  = threadIdx.x;
  const int lane = tid & 31;
  const int wave = tid >> 5;
  const int b    = blockIdx.y;
  const int ts   = blockIdx.x * TILE;

  // Fold BN into y*scale + shift; stage constants in LDS.
  if (tid < 128) {
    const int l = tid >> 5, c = tid & 31;
    const float s = p.g[l][c] * rsqrtf(p.rv[l][c] + 1e-5f);
    cbias[l][c]  = p.b[l][c];
    cscale[l][c] = s;
    cshift[l][c] = p.bb[l][c] - p.rm[l][c] * s;
  }
  if (tid < 16) { fb1s[tid] = p.fb1[tid]; fw2s[tid] = p.fw2[tid]; }
  if (tid == 0) fb2s[0] = p.fb2[0];

  // Stage input tile with halo 7: xbuf[c][i] = x[b][c][ts-7+i], zero-padded.
  #pragma unroll
  for (int c = 0; c < 4; ++c) {
    for (int i = tid; i < STRIDE; i += 256) {
      const int gpos = ts - 7 + i;
      float v = 0.0f;
      if ((unsigned)gpos < (unsigned)SLEN) v = p.x[(b * 4 + c) * SLEN + gpos];
      xbuf[c * STRIDE + i] = v;
    }
  }
  stage_weights<4, 3>(p.w[0], wstage, tid);
  __syncthreads();

  // Layer 0: 4->32, K=3.  Output covers global [ts-6, ts+134).
  conv_layer<4, 3, 9>(wstage, xbuf, bufA, cbias[0], cscale[0], cshift[0],
                      lane, wave, ts - 6);
  __syncthreads();
  stage_weights<32, 3>(p.w[1], wstage, tid);
  __syncthreads();
  // Layer 1: 32->32, K=3. Covers [ts-5, ts+133).
  conv_layer<32, 3, 9>(wstage, bufA, bufB, cbias[1], cscale[1], cshift[1],
                       lane, wave, ts - 5);
  __syncthreads();
  stage_weights<32, 5>(p.w[2], wstage, tid);
  __syncthreads();
  // Layer 2: 32->32, K=5. Covers [ts-3, ts+131).
  conv_layer<32, 5, 9>(wstage, bufB, bufA, cbias[2], cscale[2], cshift[2],
                       lane, wave, ts - 3);
  __syncthreads();
  stage_weights<32, 7>(p.w[3], wstage, tid);
  __syncthreads();
  // Layer 3: 32->32, K=7. Covers [ts, ts+128).
  conv_layer<32, 7, 8>(wstage, bufA, bufB, cbias[3], cscale[3], cshift[3],
                       lane, wave, ts);
  __syncthreads();

  // Stage fw1 [16][32] in fragment order (single m-tile, KW=1 so kc == cin).
  for (int o = tid; o < 512; o += 256) {
    const int j     = o & 1;
    const int lane2 = (o >> 1) & 31;
    const int ks    = o >> 6;
    const int kc    = ks * 4 + 2 * (lane2 >> 4) + j;
    wstage[o] = p.fw1[(lane2 & 15) * 32 + kc];
  }
  __syncthreads();

  // MLP head: h = relu(act^T fw1^T + fb1); out = h fw2^T + fb2.
  const int n_   = lane & 15;
  const int half = lane >> 4;
  const v2f* aw  = reinterpret_cast<const v2f*>(wstage) + lane;
  for (int nt = wave; nt < 8; nt += 8) {
    const int nb = nt * 16;
    const float* b2 = bufB + 2 * half * STRIDE + nb + n_;
    v8f acc = {};
    #pragma unroll
    for (int ks = 0; ks < 8; ++ks) {
      v2f bf;
      bf.x = b2[(ks * 4 + 0) * STRIDE];
      bf.y = b2[(ks * 4 + 1) * STRIDE];
      const v2f a = aw[ks * 32];
      acc = __builtin_amdgcn_wmma_f32_16x16x4_f32(false, a, false, bf,
                                                  (short)0, acc, false, false);
    }
    // 16->1: each lane reduces its 8 rows, then combine halves across lane^16.
    const float* fb = fb1s + 8 * half;
    const float* fw = fw2s + 8 * half;
    float partial = 0.0f;
    #pragma unroll
    for (int i = 0; i < 8; ++i) {
      const float h = fmaxf(acc[i] + fb[i], 0.0f);
      partial += h * fw[i];
    }
    const float tot = partial + __shfl_xor(partial, 16, 32);
    if (half == 0) p.out[b * SLEN + ts + nb + n_] = tot + fb2s[0];
  }
}

extern "C" void kernel_launch(void* const* d_in, const int* in_sizes, int n_in,
                              void* d_out, int out_size, void* d_ws, size_t ws_size,
                              hipStream_t stream) {
  (void)in_sizes; (void)n_in; (void)out_size; (void)d_ws; (void)ws_size;
  Params p;
  p.x = (const float*)d_in[0];
  for (int l = 0; l < 4; ++l) {
    p.w[l]  = (const float*)d_in[1 + 6 * l + 0];
    p.b[l]  = (const float*)d_in[1 + 6 * l + 1];
    p.g[l]  = (const float*)d_in[1 + 6 * l + 2];
    p.bb[l] = (const float*)d_in[1 + 6 * l + 3];
    p.rm[l] = (const float*)d_in[1 + 6 * l + 4];
    p.rv[l] = (const float*)d_in[1 + 6 * l + 5];
  }
  p.fw1 = (const float*)d_in[25];
  p.fb1 = (const float*)d_in[26];
  p.fw2 = (const float*)d_in[27];
  p.fb2 = (const float*)d_in[28];
  p.out = (float*)d_out;

  dim3 grid(SLEN / TILE, BATCH);
  dnashape_fused_kernel<<<grid, 256, 0, stream>>>(p);
}